// AttentionPoolingPerImage_55671366091486
// MI455X (gfx1250) — compile-verified
//
#include <hip/hip_runtime.h>
#include <hip/hip_bf16.h>
#include <math.h>

// ---------------------------------------------------------------------------
// AttentionPoolingPerImage, MI455X (gfx1250, wave32).
//
// Algebra: softmax weights sum to 1 per segment, so
//   pooled[n] = Ahat[n] @ W_value + b_value,   Ahat[n] = sum_i attn_w[i]*emb[i]
//   r_images[n] = pooled[n] . w_out + b_out
// b_attn cancels (softmax shift invariance). The 1M-row GEMM disappears;
// the kernel is HBM-bound: 2 passes over emb (2 x 256 MB) ~= 22 us @ 23.3 TB/s.
// The surviving matmul (4096x64 @ 64x64) is done with V_WMMA_F32_16X16X4_F32.
// ---------------------------------------------------------------------------

#define EMB_DIM    64
#define N_IMAGES   4096
#define SEG_WAVES  8           // 256 threads per segment block
#define SCORE_CAP  1024        // LDS score cache; Poisson(256) => overflow ~never

typedef float v2f __attribute__((ext_vector_type(2)));
typedef float v8f __attribute__((ext_vector_type(8)));

__device__ __forceinline__ float wave_reduce_sum(float v) {
  #pragma unroll
  for (int m = 16; m >= 1; m >>= 1) v += __shfl_xor(v, m, 32);
  return v;   // uniform across all 32 lanes
}

__device__ __forceinline__ int lower_bound_i32(const int* __restrict__ a, int n, int key) {
  int lo = 0, hi = n;
  while (lo < hi) {
    int mid = (lo + hi) >> 1;
    if (a[mid] < key) lo = mid + 1; else hi = mid;
  }
  return lo;
}

// ---------------------------------------------------------------------------
// Kernel 1: per-image segment softmax pooling of embeddings.
// One block (8 wave32) per image. ids are sorted -> binary search bounds.
// Pass 1: per-row score dot (lane owns 2 columns, coalesced 256B/row/wave),
//         cache scores in LDS, track segment max.
// Pass 2: ex = exp(score - max); accumulate sum(ex) and sum(ex * emb).
// Output: Ahat[n][64] = sum(ex*emb)/sum(ex)  (softmax-weighted embedding).
// ---------------------------------------------------------------------------
__global__ __launch_bounds__(256)
void seg_softmax_pool_kernel(const float* __restrict__ emb,
                             const int*   __restrict__ ids,
                             const float* __restrict__ w_attn,
                             float*       __restrict__ Ahat,
                             int Btot) {
  __shared__ float sc[SCORE_CAP];
  __shared__ float Apart[SEG_WAVES][EMB_DIM];
  __shared__ float Mpart[SEG_WAVES];
  __shared__ float Spart[SEG_WAVES];
  __shared__ int   seg_lo_sh, seg_len_sh;
  __shared__ float m_sh, s_sh;

  const int n    = blockIdx.x;
  const int tid  = threadIdx.x;
  const int lane = tid & 31;
  const int wv   = tid >> 5;

  if (tid == 0) {
    int lo = lower_bound_i32(ids, Btot, n);
    int hi = lower_bound_i32(ids, Btot, n + 1);
    seg_lo_sh = lo;
    seg_len_sh = hi - lo;
  }
  __syncthreads();
  const int lo  = seg_lo_sh;
  const int len = seg_len_sh;

  const float2  wa   = ((const float2*)w_attn)[lane];   // lane owns cols 2l,2l+1
  const float2* emb2 = (const float2*)emb;

  // ---- pass 1: scores + segment max -------------------------------------
  float mymax = -3.402823466e38f;
  for (int i = wv; i < len; i += SEG_WAVES) {
    float2 e = emb2[(size_t)(lo + i) * 32 + lane];
    float  p = wave_reduce_sum(e.x * wa.x + e.y * wa.y);   // uniform score
    if (i < SCORE_CAP && lane == 0) sc[i] = p;
    mymax = fmaxf(mymax, p);
  }
  if (lane == 0) Mpart[wv] = mymax;
  __syncthreads();
  if (tid == 0) {
    float m = Mpart[0];
    #pragma unroll
    for (int w = 1; w < SEG_WAVES; ++w) m = fmaxf(m, Mpart[w]);
    m_sh = m;
  }
  __syncthreads();
  const float m = m_sh;

  // ---- pass 2: exp-weighted accumulation --------------------------------
  float ax = 0.f, ay = 0.f, sacc = 0.f;
  for (int i = wv; i < len; i += SEG_WAVES) {
    float2 e = emb2[(size_t)(lo + i) * 32 + lane];
    float sci;
    if (i < SCORE_CAP) {
      sci = sc[i];                                   // LDS broadcast read
    } else {
      sci = wave_reduce_sum(e.x * wa.x + e.y * wa.y);
    }
    float ex = __expf(sci - m);                      // uniform across wave
    ax   += ex * e.x;
    ay   += ex * e.y;
    sacc += ex;
  }
  Apart[wv][2 * lane]     = ax;
  Apart[wv][2 * lane + 1] = ay;
  if (lane == 0) Spart[wv] = sacc;
  __syncthreads();
  if (tid == 0) {
    float s = 0.f;
    #pragma unroll
    for (int w = 0; w < SEG_WAVES; ++w) s += Spart[w];
    s_sh = s;
  }
  __syncthreads();
  if (tid < EMB_DIM) {
    float a = 0.f;
    #pragma unroll
    for (int w = 0; w < SEG_WAVES; ++w) a += Apart[w][tid];
    float s = s_sh;
    Ahat[(size_t)n * EMB_DIM + tid] = (len > 0 && s > 0.f) ? (a / s) : 0.f;
  }
}

// ---------------------------------------------------------------------------
// Kernel 2: r_images = (Ahat @ W_value + b_value) . w_out + b_out
// One wave per 16 rows. WMMA f32 16x16x4: A tile = 16 rows x 4 K,
// 4 N-tiles (64 cols) x 16 K-steps = 64 v_wmma_f32_16x16x4_f32 per wave.
// A layout (ISA 7.12.2): lanes 0-15 hold K={0,1}, lanes 16-31 hold K={2,3}.
// B layout: N striped across lanes within a VGPR, same K split per half.
// C/D layout: VGPR v -> M=v (lanes 0-15) / M=v+8 (lanes 16-31), N=lane&15.
// ---------------------------------------------------------------------------
__global__ __launch_bounds__(32)
void pooled_gemm_wmma_kernel(const float* __restrict__ Ahat,
                             const float* __restrict__ Wv,
                             const float* __restrict__ bv,
                             const float* __restrict__ w_out,
                             const float* __restrict__ b_out,
                             float*       __restrict__ r_img) {
  const int lane    = threadIdx.x;
  const int half    = lane >> 4;          // 0: K+{0,1}, 1: K+{2,3}
  const int l15     = lane & 15;
  const int rowBase = blockIdx.x * 16;

  v8f acc[4];
  #pragma unroll
  for (int t = 0; t < 4; ++t)
    #pragma unroll
    for (int v = 0; v < 8; ++v) acc[t][v] = 0.f;

  #pragma unroll 4
  for (int kk = 0; kk < 16; ++kk) {
    const int k0 = kk * 4 + half * 2;
    v2f a;
    a.x = Ahat[(size_t)(rowBase + l15) * EMB_DIM + k0];
    a.y = Ahat[(size_t)(rowBase + l15) * EMB_DIM + k0 + 1];
    #pragma unroll
    for (int t = 0; t < 4; ++t) {
      v2f b;
      b.x = Wv[(size_t)k0 * EMB_DIM + t * 16 + l15];
      b.y = Wv[(size_t)(k0 + 1) * EMB_DIM + t * 16 + l15];
      acc[t] = __builtin_amdgcn_wmma_f32_16x16x4_f32(
          /*neg_a=*/false, a, /*neg_b=*/false, b,
          /*c_mod=*/(short)0, acc[t], /*reuse_a=*/false, /*reuse_b=*/false);
    }
  }

  // Fused epilogue: pooled = acc + b_value; r = pooled . w_out + b_out.
  float wvec[4], bvv[4];
  #pragma unroll
  for (int t = 0; t < 4; ++t) {
    const int h = t * 16 + l15;
    wvec[t] = w_out[h];
    bvv[t]  = bv[h];
  }
  const float bo = b_out[0];

  #pragma unroll
  for (int v = 0; v < 8; ++v) {
    float r = 0.f;
    #pragma unroll
    for (int t = 0; t < 4; ++t) r += (acc[t][v] + bvv[t]) * wvec[t];
    // reduce across the 16 lanes of each half (masks < 16 stay in-half)
    #pragma unroll
    for (int msk = 8; msk >= 1; msk >>= 1) r += __shfl_xor(r, msk, 32);
    if (l15 == 0) r_img[rowBase + half * 8 + v] = r + bo;
  }
}

// ---------------------------------------------------------------------------
// Kernel 3: r_reflections gather + unique_ids. r_images already lives in
// d_out[B .. B+N); gather reads it (prior kernel on same stream ordered).
// ---------------------------------------------------------------------------
__global__ __launch_bounds__(256)
void finalize_kernel(const int* __restrict__ ids,
                     float*     __restrict__ out,
                     int Btot, int Nimg) {
  const int i = blockIdx.x * blockDim.x + threadIdx.x;
  const float* __restrict__ r_img = out + Btot;
  if (i < Btot) out[i] = r_img[ids[i]];
  if (i < Nimg) out[(size_t)Btot + Nimg + i] = (float)i;
}

// ---------------------------------------------------------------------------
extern "C" void kernel_launch(void* const* d_in, const int* in_sizes, int n_in,
                              void* d_out, int out_size, void* d_ws, size_t ws_size,
                              hipStream_t stream) {
  const float* emb     = (const float*)d_in[0];
  const int*   ids     = (const int*)  d_in[1];
  const float* w_attn  = (const float*)d_in[2];
  // d_in[3] = b_attn: unused, softmax is shift-invariant.
  const float* w_value = (const float*)d_in[4];
  const float* b_value = (const float*)d_in[5];
  const float* w_out   = (const float*)d_in[6];
  const float* b_out   = (const float*)d_in[7];
  float*       out     = (float*)d_out;

  const int Btot = in_sizes[1];           // number of reflections (1048576)
  float* Ahat = (float*)d_ws;             // N_IMAGES x 64 floats = 1 MB scratch

  seg_softmax_pool_kernel<<<N_IMAGES, 256, 0, stream>>>(emb, ids, w_attn, Ahat, Btot);
  pooled_gemm_wmma_kernel<<<N_IMAGES / 16, 32, 0, stream>>>(
      Ahat, w_value, b_value, w_out, b_out, out + Btot);
  finalize_kernel<<<(Btot + 255) / 256, 256, 0, stream>>>(ids, out, Btot, N_IMAGES);
}